// CrossLayerMemoryManager_13932873908438
// MI455X (gfx1250) — compile-verified
//
#include <hip/hip_runtime.h>
#include <math.h>

// Problem constants (match reference)
#define B_    16
#define S_    2048
#define H_    4096
#define M_    128
#define TOPK_ 16
#define SCHUNK 8   // S split for the pooled reduction (deterministic two-stage)

typedef float v2f __attribute__((ext_vector_type(2)));
typedef float v4f __attribute__((ext_vector_type(4)));
typedef float v8f __attribute__((ext_vector_type(8)));

// ---------------------------------------------------------------------------
// K1a: partial sums over S.  grid (H/1024, B, SCHUNK), block 256.
// Partials are written into d_out's combined region (scratch until K6).
// ---------------------------------------------------------------------------
__global__ void pool_partial_kernel(const float* __restrict__ hs,
                                    float* __restrict__ part) {
  const int tid = threadIdx.x;
  const int hc  = blockIdx.x;          // 0..3   (1024-col chunk)
  const int b   = blockIdx.y;          // 0..15
  const int sc  = blockIdx.z;          // 0..SCHUNK-1
  const int h   = hc * 1024 + tid * 4;
  const v4f* src = (const v4f*)(hs + ((size_t)b * S_ + (size_t)sc * (S_ / SCHUNK)) * H_ + h);
  v4f acc = {0.f, 0.f, 0.f, 0.f};
  for (int s = 0; s < S_ / SCHUNK; ++s) {
    v4f v = __builtin_nontemporal_load(src);   // streaming: 512MB >> L2
    src += H_ / 4;
    acc += v;
  }
  *(v4f*)(part + ((size_t)sc * B_ + b) * H_ + h) = acc;
}

// K1b: reduce SCHUNK partials -> pooled [B,H] (mean).  grid 64, block 256.
__global__ void pool_reduce_kernel(const float* __restrict__ part,
                                   float* __restrict__ pooled) {
  const size_t o = ((size_t)blockIdx.x * blockDim.x + threadIdx.x) * 4;
  v4f acc = {0.f, 0.f, 0.f, 0.f};
  for (int c = 0; c < SCHUNK; ++c)
    acc += *(const v4f*)(part + (size_t)c * B_ * H_ + o);
  acc *= (1.0f / (float)S_);
  *(v4f*)(pooled + o) = acc;
}

// ---------------------------------------------------------------------------
// K2: logits = pooled @ W.T + b for alloc & retr, via V_WMMA_F32_16X16X4_F32.
// grid (8 n-tiles, 2 matrices), block 32 (one wave; EXEC all ones).
// A layout (ISA 7.12.2, 32-bit A 16x4): lane&15 = row M, lanes>=16 hold K+2.
// B layout (4x16): VGPR0 lanes0-15 = K row 0 (N=lane), lanes16-31 = K row 2.
// ---------------------------------------------------------------------------
__global__ void logits_wmma_kernel(const float* __restrict__ pooled,
                                   const float* __restrict__ allocW,
                                   const float* __restrict__ allocB,
                                   const float* __restrict__ retrW,
                                   const float* __restrict__ retrB,
                                   float* __restrict__ allocL,
                                   float* __restrict__ retrL) {
  const int lane = threadIdx.x;
  const int half = lane >> 4;          // 0 or 1
  const int l16  = lane & 15;
  const int nbase = blockIdx.x * 16;   // memory-slot tile
  const int which = blockIdx.y;        // 0=alloc, 1=retr
  const float* W    = which ? retrW : allocW;
  const float* bias = which ? retrB : allocB;
  float*       out  = which ? retrL : allocL;

  const float* aPtr = pooled + (size_t)l16 * H_ + 2 * half;            // A[M][k]
  const float* bPtr = W + (size_t)(nbase + l16) * H_ + 2 * half;       // B[k][N]=W[N][k]

  v8f c = {};
  for (int k0 = 0; k0 < H_; k0 += 4) {
    v2f a = *(const v2f*)(aPtr + k0);
    v2f b = *(const v2f*)(bPtr + k0);
    c = __builtin_amdgcn_wmma_f32_16x16x4_f32(false, a, false, b, (short)0, c, false, false);
  }
  // C/D layout: VGPR r -> (M=r, N=lane) for lanes 0-15, (M=r+8, N=lane-16) else
  for (int r = 0; r < 8; ++r) {
    const int row = r + 8 * half;        // batch row 0..15
    const int col = nbase + l16;         // memory slot
    out[row * M_ + col] = c[r] + bias[col];
  }
}

// K3: softmax over M=128 per (batch, matrix) row. grid 32, block 128, in-place.
__global__ void softmax128_kernel(float* __restrict__ allocL,
                                  float* __restrict__ retrL) {
  __shared__ float s[128];
  const int t   = threadIdx.x;
  const int row = blockIdx.x & 15;
  float* buf = (blockIdx.x >> 4) ? retrL : allocL;
  const float x = buf[row * M_ + t];
  s[t] = x; __syncthreads();
  for (int o = 64; o > 0; o >>= 1) { if (t < o) s[t] = fmaxf(s[t], s[t + o]); __syncthreads(); }
  const float mx = s[0]; __syncthreads();
  const float e = __expf(x - mx);
  s[t] = e; __syncthreads();
  for (int o = 64; o > 0; o >>= 1) { if (t < o) s[t] += s[t + o]; __syncthreads(); }
  buf[row * M_ + t] = e / s[0];
}

// ---------------------------------------------------------------------------
// K4: retrieved = retr_w @ memory_bank, fused epilogue:
//     upd_in = pooled + scale*retrieved  (== mean_S(combined), exactly)
// grid 256 (H/16 tiles), block 32.
// ---------------------------------------------------------------------------
__global__ void retrieve_wmma_kernel(const float* __restrict__ retrWsoft,
                                     const float* __restrict__ memBank,
                                     const float* __restrict__ pooled,
                                     const float* __restrict__ scales,
                                     const int*   __restrict__ lidx,
                                     float* __restrict__ retrieved,
                                     float* __restrict__ updIn) {
  const int lane = threadIdx.x, half = lane >> 4, l16 = lane & 15;
  const int nbase = blockIdx.x * 16;
  const float* aPtr = retrWsoft + (size_t)l16 * M_ + 2 * half;
  v8f c = {};
  for (int k0 = 0; k0 < M_; k0 += 4) {
    v2f a = *(const v2f*)(aPtr + k0);
    v2f b;
    b.x = memBank[(size_t)(k0 + 2 * half)     * H_ + nbase + l16];
    b.y = memBank[(size_t)(k0 + 2 * half + 1) * H_ + nbase + l16];
    c = __builtin_amdgcn_wmma_f32_16x16x4_f32(false, a, false, b, (short)0, c, false, false);
  }
  const float sc = scales[*lidx];
  for (int r = 0; r < 8; ++r) {
    const int row = r + 8 * half;
    const int col = nbase + l16;
    const float v = c[r];
    retrieved[(size_t)row * H_ + col] = v;
    updIn[(size_t)row * H_ + col] = pooled[(size_t)row * H_ + col] + sc * v;
  }
}

// K5: update_signals = sigmoid(upd_in @ upd_W.T + upd_b). grid 256, block 32.
__global__ void update_wmma_kernel(const float* __restrict__ updIn,
                                   const float* __restrict__ updW,
                                   const float* __restrict__ updB,
                                   float* __restrict__ updSig) {
  const int lane = threadIdx.x, half = lane >> 4, l16 = lane & 15;
  const int nbase = blockIdx.x * 16;
  const float* aPtr = updIn + (size_t)l16 * H_ + 2 * half;
  const float* bPtr = updW + (size_t)(nbase + l16) * H_ + 2 * half;   // B[k][N]=updW[N][k]
  v8f c = {};
  for (int k0 = 0; k0 < H_; k0 += 4) {
    v2f a = *(const v2f*)(aPtr + k0);
    v2f b = *(const v2f*)(bPtr + k0);
    c = __builtin_amdgcn_wmma_f32_16x16x4_f32(false, a, false, b, (short)0, c, false, false);
  }
  for (int r = 0; r < 8; ++r) {
    const int row = r + 8 * half;
    const int col = nbase + l16;
    const float z = c[r] + updB[col];
    updSig[(size_t)row * H_ + col] = 1.0f / (1.0f + __expf(-z));
  }
}

// K6: combined = hidden + scale*retrieved (broadcast). grid 131072, block 256.
__global__ void combine_kernel(const float* __restrict__ hs,
                               const float* __restrict__ retrieved,
                               const float* __restrict__ scales,
                               const int*   __restrict__ lidx,
                               float* __restrict__ out) {
  const size_t i4 = (size_t)blockIdx.x * blockDim.x + threadIdx.x;
  const size_t e  = i4 * 4;
  const int h = (int)(e & (size_t)(H_ - 1));
  const int b = (int)(e >> 23);                 // S_*H_ = 2^23
  const float sc = scales[*lidx];
  v4f hv = __builtin_nontemporal_load((const v4f*)(hs + e));
  v4f rv = *(const v4f*)(retrieved + (size_t)b * H_ + h);
  v4f o = hv + sc * rv;
  __builtin_nontemporal_store(o, (v4f*)(out + e));
}

// K7: rank-based top-16 per batch row; last-writer-wins. grid 1, block 128.
// Rank order: value desc, index asc (matches jax.lax.top_k tie-breaking).
__global__ void topk_lastwriter_kernel(const float* __restrict__ allocWsoft,
                                       int* __restrict__ lw) {
  __shared__ float s[M_];
  const int m = threadIdx.x;
  int lwLocal = -1;
  for (int b = 0; b < B_; ++b) {
    s[m] = allocWsoft[b * M_ + m];
    __syncthreads();
    const float mine = s[m];
    int cnt = 0;
    for (int j = 0; j < M_; ++j) {
      const float vj = s[j];
      cnt += (vj > mine) || (vj == mine && j < m);
    }
    if (cnt < TOPK_) lwLocal = b;   // ascending b => last write wins
    __syncthreads();
  }
  lw[m] = lwLocal;
}

// K8: new_memory_bank row copy. grid 128, block 256.
__global__ void bank_write_kernel(const float* __restrict__ memBank,
                                  const float* __restrict__ updSig,
                                  const int*   __restrict__ lw,
                                  float* __restrict__ outBank) {
  const int m = blockIdx.x;
  const int w = lw[m];
  const float* src = (w >= 0) ? (updSig + (size_t)w * H_) : (memBank + (size_t)m * H_);
  float* dst = outBank + (size_t)m * H_;
  for (int h = threadIdx.x * 4; h < H_; h += blockDim.x * 4)
    *(v4f*)(dst + h) = *(const v4f*)(src + h);
}

// ---------------------------------------------------------------------------
extern "C" void kernel_launch(void* const* d_in, const int* in_sizes, int n_in,
                              void* d_out, int out_size, void* d_ws, size_t ws_size,
                              hipStream_t stream) {
  const float* hs      = (const float*)d_in[0];  // [B,S,H]
  const float* memBank = (const float*)d_in[1];  // [M,H]
  const float* allocW  = (const float*)d_in[2];  // [M,H]
  const float* allocB  = (const float*)d_in[3];  // [M]
  const float* retrW   = (const float*)d_in[4];  // [M,H]
  const float* retrB   = (const float*)d_in[5];  // [M]
  const float* updW    = (const float*)d_in[6];  // [H,H]
  const float* updB    = (const float*)d_in[7];  // [H]
  const float* scales  = (const float*)d_in[8];  // [L]
  const int*   lidx    = (const int*)  d_in[9];  // scalar

  float* outCombined = (float*)d_out;                         // [B,S,H]
  float* outBank     = outCombined + (size_t)B_ * S_ * H_;    // [M,H]

  // Workspace layout (floats); ~1.05 MB total.
  float* ws       = (float*)d_ws;
  float* pooled   = ws;                       // 16*4096
  float* allocL   = pooled   + B_ * H_;       // 16*128
  float* retrL    = allocL   + B_ * M_;       // 16*128
  float* retrieved= retrL    + B_ * M_;       // 16*4096
  float* updIn    = retrieved+ B_ * H_;       // 16*4096
  float* updSig   = updIn    + B_ * H_;       // 16*4096
  int*   lw       = (int*)(updSig + B_ * H_); // 128

  // Pooled-reduction partials live in d_out's combined region (scratch until K6).
  float* part = outCombined;                  // SCHUNK*B*H floats = 2 MB

  pool_partial_kernel<<<dim3(H_ / 1024, B_, SCHUNK), 256, 0, stream>>>(hs, part);
  pool_reduce_kernel<<<(B_ * H_) / (256 * 4), 256, 0, stream>>>(part, pooled);

  logits_wmma_kernel<<<dim3(M_ / 16, 2), 32, 0, stream>>>(
      pooled, allocW, allocB, retrW, retrB, allocL, retrL);
  softmax128_kernel<<<32, 128, 0, stream>>>(allocL, retrL);

  retrieve_wmma_kernel<<<H_ / 16, 32, 0, stream>>>(
      retrL, memBank, pooled, scales, lidx, retrieved, updIn);
  update_wmma_kernel<<<H_ / 16, 32, 0, stream>>>(updIn, updW, updB, updSig);

  combine_kernel<<<(size_t)B_ * S_ * H_ / (256 * 4), 256, 0, stream>>>(
      hs, retrieved, scales, lidx, outCombined);

  topk_lastwriter_kernel<<<1, 128, 0, stream>>>(allocL, lw);
  bank_write_kernel<<<M_, 256, 0, stream>>>(memBank, updSig, lw, outBank);
}